// SDE_56813827392342
// MI455X (gfx1250) — compile-verified
//
#include <hip/hip_runtime.h>
#include <math.h>

// ---------------------------------------------------------------------------
// Moment-matched SDE forward (stochastic interpolant), P=6, sigma=0.5.
// Per step k:
//   IR(k)        : Sum I_k^m*Idot, Sum I_k1^j, Sum I_k1^m*Idot  (18 sums)
//   solve1       : eta  = G(x)^-1 rhs_dt          (6x6, 1 thread)
//   UR predictor : y = x + h*gradphi(x)@eta + sqrt(2h)*sigma*z ; S_m(y) m=0..10
//   solve2       : theta= G(y)^-1 (bpe - moments(y)) ; theta/dH/bpe rows
//   UR corrector : x' = y + gradphi(y)@theta ; S_m(x') (feeds next step's G(x))
// Reductions: per-thread regs -> wave32 shfl butterfly -> 8x32 LDS tile ->
// per-block partials[NBLK][32] -> single-wave WMMA reducer (A = ones,
// V_WMMA_F32_16X16X4_F32) collapsing all block partials into 32 exact f32 sums.
// Streaming passes use float4 (global_load_b128) for bandwidth efficiency.
// ---------------------------------------------------------------------------

typedef __attribute__((ext_vector_type(2))) float v2f;
typedef __attribute__((ext_vector_type(8))) float v8f;

#define BLK   256
#define NWAVE (BLK / 32)
#define NBLK  1024
#define NACC  32
#define PI_2  1.57079632679489662f

// ---------------- deterministic counter-based Gaussian (splitmix64 + BoxMuller)
__device__ __forceinline__ float gauss_kn(unsigned k, unsigned i) {
  unsigned long long s = (((unsigned long long)k) << 32) ^ (unsigned long long)i;
  s += 0x9E3779B97F4A7C15ull;
  unsigned long long z = s;
  z = (z ^ (z >> 30)) * 0xBF58476D1CE4E5B9ull;
  z = (z ^ (z >> 27)) * 0x94D049BB133111EBull;
  z ^= z >> 31;
  unsigned lo = (unsigned)z, hi = (unsigned)(z >> 32);
  float u1 = (float)(lo >> 8) * (1.0f / 16777216.0f) + 5.96e-8f;  // (0,1]
  float u2 = (float)(hi >> 8) * (1.0f / 16777216.0f);
  float r  = __builtin_sqrtf(-2.0f * __logf(u1));
  return r * __cosf(6.28318530718f * u2);
}

// ---------------- tiny 6x6 Gaussian elimination with partial pivoting
__device__ __forceinline__ void solve6(float A[6][7], float x[6]) {
  for (int c = 0; c < 6; ++c) {
    int p = c; float best = fabsf(A[c][c]);
    for (int r = c + 1; r < 6; ++r) { float v = fabsf(A[r][c]); if (v > best) { best = v; p = r; } }
    if (p != c) for (int q = c; q < 7; ++q) { float tmp = A[c][q]; A[c][q] = A[p][q]; A[p][q] = tmp; }
    float inv = 1.0f / A[c][c];
    for (int r = c + 1; r < 6; ++r) {
      float f = A[r][c] * inv;
      for (int q = c; q < 7; ++q) A[r][q] -= f * A[c][q];
    }
  }
  for (int r = 5; r >= 0; --r) {
    float acc = A[r][6];
    for (int q = r + 1; q < 6; ++q) acc -= A[r][q] * x[q];
    x[r] = acc / A[r][r];
  }
}

__global__ void zero8_kernel(float* p) {
  if (threadIdx.x < 8) p[threadIdx.x] = 0.0f;
}

// ---------------- block reduction: wave32 shfl butterfly -> 8x32 LDS -> store
template <int NA>
__device__ __forceinline__ void block_reduce_store(float* acc, float* __restrict__ partials) {
  __shared__ float wred[NWAVE][NACC];
  const int tid  = threadIdx.x;
  const int wid  = tid >> 5;
  const int lane = tid & 31;
  #pragma unroll
  for (int m = 0; m < NA; ++m) {
    float a = acc[m];
    #pragma unroll
    for (int off = 16; off > 0; off >>= 1) a += __shfl_down(a, off, 32);
    if (lane == 0) wred[wid][m] = a;
  }
  __syncthreads();
  if (tid < NACC) {
    float s = 0.0f;
    if (tid < NA) {
      #pragma unroll
      for (int w = 0; w < NWAVE; ++w) s += wred[w][tid];
    }
    partials[blockIdx.x * NACC + tid] = s;   // zero-filled for unused slots
  }
}

// ---------------- update + power-sum reduction (float4 main loop):
// v = u + s * sum_{m=0..5}(m+1)*coef[m]*u^m  (+ sqrt(2h)*sigma*z if mode)
// accumulates S_m(v), m = 0..10
__global__ __launch_bounds__(BLK)
void ur_kernel(const float* __restrict__ u, float* __restrict__ v,
               const float* __restrict__ coef, const float* __restrict__ t,
               int k, int mode, int N, float* __restrict__ partials) {
  float s = 1.0f, ns = 0.0f;
  if (mode) {
    float h = t[k + 1] - t[k];
    s = h;
    ns = __builtin_sqrtf(2.0f * h) * 0.5f;  // SIGMA = 0.5
  }
  const float d0 = 1.0f * coef[0], d1 = 2.0f * coef[1], d2 = 3.0f * coef[2];
  const float d3 = 4.0f * coef[3], d4 = 5.0f * coef[4], d5 = 6.0f * coef[5];

  float acc[11];
  #pragma unroll
  for (int m = 0; m < 11; ++m) acc[m] = 0.0f;

  const int gid    = blockIdx.x * blockDim.x + threadIdx.x;
  const int stride = gridDim.x * blockDim.x;
  const int N4 = N >> 2;
  const float4* __restrict__ u4 = (const float4*)u;
  float4* __restrict__ v4 = (float4*)v;

  for (int i = gid; i < N4; i += stride) {
    const float4 uu = u4[i];
    float in[4]  = {uu.x, uu.y, uu.z, uu.w};
    float outv[4];
    #pragma unroll
    for (int c = 0; c < 4; ++c) {
      const float u0 = in[c];
      float drift = fmaf(u0, d5, d4);
      drift = fmaf(u0, drift, d3);
      drift = fmaf(u0, drift, d2);
      drift = fmaf(u0, drift, d1);
      drift = fmaf(u0, drift, d0);
      float vv = fmaf(s, drift, u0);
      if (mode) vv = fmaf(ns, gauss_kn((unsigned)k, (unsigned)(4 * i + c)), vv);
      outv[c] = vv;
      acc[0] += 1.0f;
      float pw = vv;
      acc[1] += pw;
      #pragma unroll
      for (int m = 2; m <= 10; ++m) { pw *= vv; acc[m] += pw; }
    }
    float4 vo; vo.x = outv[0]; vo.y = outv[1]; vo.z = outv[2]; vo.w = outv[3];
    v4[i] = vo;
  }
  // scalar tail (N not divisible by 4)
  for (int i = (N & ~3) + gid; i < N; i += stride) {
    const float u0 = u[i];
    float drift = fmaf(u0, d5, d4);
    drift = fmaf(u0, drift, d3);
    drift = fmaf(u0, drift, d2);
    drift = fmaf(u0, drift, d1);
    drift = fmaf(u0, drift, d0);
    float vv = fmaf(s, drift, u0);
    if (mode) vv = fmaf(ns, gauss_kn((unsigned)k, (unsigned)i), vv);
    v[i] = vv;
    acc[0] += 1.0f;
    float pw = vv;
    acc[1] += pw;
    #pragma unroll
    for (int m = 2; m <= 10; ++m) { pw *= vv; acc[m] += pw; }
  }
  block_reduce_store<11>(acc, partials);
}

// ---------------- interpolant reduction (float4 main loop):
// slots 0..5 : Sum I_k^m * Idot    (rhs_dt numerators)
// slots 6..11: Sum I_k1^j, j=1..6  (bpe numerators)
// slots 12..17: Sum I_k1^m * Idot  (dtphi numerators)
__global__ __launch_bounds__(BLK)
void ir_kernel(const float* __restrict__ x0, const float* __restrict__ x1,
               const float* __restrict__ t, int k, int N,
               float* __restrict__ partials) {
  const float tk = t[k], tk1 = t[k + 1];
  const float a  = cosf(PI_2 * tk),  b  = sinf(PI_2 * tk);
  const float a1 = cosf(PI_2 * tk1), b1 = sinf(PI_2 * tk1);

  float acc[18];
  #pragma unroll
  for (int m = 0; m < 18; ++m) acc[m] = 0.0f;

  const int gid    = blockIdx.x * blockDim.x + threadIdx.x;
  const int stride = gridDim.x * blockDim.x;
  const int N4 = N >> 2;
  const float4* __restrict__ x04 = (const float4*)x0;
  const float4* __restrict__ x14 = (const float4*)x1;

  for (int i = gid; i < N4; i += stride) {
    const float4 a0 = x04[i];
    const float4 a1v = x14[i];
    float X0s[4] = {a0.x, a0.y, a0.z, a0.w};
    float X1s[4] = {a1v.x, a1v.y, a1v.z, a1v.w};
    #pragma unroll
    for (int c = 0; c < 4; ++c) {
      const float X0 = X0s[c], X1 = X1s[c];
      const float I  = fmaf(a, X0, b * X1);
      const float Id = PI_2 * fmaf(a, X1, -b * X0);
      const float I1 = fmaf(a1, X0, b1 * X1);
      float pw = 1.0f;
      #pragma unroll
      for (int m = 0; m < 6; ++m) { acc[m] += pw * Id; pw *= I; }
      pw = I1;
      #pragma unroll
      for (int j = 1; j <= 6; ++j) { acc[5 + j] += pw; pw *= I1; }
      pw = 1.0f;
      #pragma unroll
      for (int m = 0; m < 6; ++m) { acc[12 + m] += pw * Id; pw *= I1; }
    }
  }
  for (int i = (N & ~3) + gid; i < N; i += stride) {
    const float X0 = x0[i], X1 = x1[i];
    const float I  = fmaf(a, X0, b * X1);
    const float Id = PI_2 * fmaf(a, X1, -b * X0);
    const float I1 = fmaf(a1, X0, b1 * X1);
    float pw = 1.0f;
    #pragma unroll
    for (int m = 0; m < 6; ++m) { acc[m] += pw * Id; pw *= I; }
    pw = I1;
    #pragma unroll
    for (int j = 1; j <= 6; ++j) { acc[5 + j] += pw; pw *= I1; }
    pw = 1.0f;
    #pragma unroll
    for (int m = 0; m < 6; ++m) { acc[12 + m] += pw * Id; pw *= I1; }
  }
  block_reduce_store<18>(acc, partials);
}

// ---------------- stage-2 reducer: one full wave32, V_WMMA_F32_16X16X4_F32.
// A = ones(16x4)  =>  D[m,n] = sum_k B[k,n]; two column halves cover 32 accums.
// f32 B 4x16 layout (mirror of A 16x4): VGPR0 = rows {K=0 | K=2},
// VGPR1 = rows {K=1 | K=3}, N striped across lanes within each half.
__global__ void wmma_reduce_kernel(const float* __restrict__ partials,
                                   float* __restrict__ sums, int nb) {
  const int lane = threadIdx.x;
  const int n  = lane & 15;
  const int kc = (lane >> 4) << 1;          // 0 for lanes 0-15, 2 for lanes 16-31
  v2f a; a.x = 1.0f; a.y = 1.0f;            // ones matrix
  v8f clo = {0.f,0.f,0.f,0.f,0.f,0.f,0.f,0.f};
  v8f chi = {0.f,0.f,0.f,0.f,0.f,0.f,0.f,0.f};
  for (int b = 0; b < nb; b += 4) {
    const float* base = partials + (size_t)(b + kc) * NACC;
    v2f blo, bhi;
    blo.x = base[n];           blo.y = base[NACC + n];
    bhi.x = base[16 + n];      bhi.y = base[NACC + 16 + n];
    clo = __builtin_amdgcn_wmma_f32_16x16x4_f32(false, a, false, blo, (short)0, clo, false, false);
    chi = __builtin_amdgcn_wmma_f32_16x16x4_f32(false, a, false, bhi, (short)0, chi, false, false);
  }
  // D row 0 lives in VGPR0, lanes 0-15 (all rows equal since A is all ones)
  if (lane < 16) { sums[n] = clo[0]; sums[16 + n] = chi[0]; }
}

// ---------------- predictor solve: eta = G(x)^-1 rhs_dt
__global__ void solve1_kernel(const float* __restrict__ xs, const float* __restrict__ irs,
                              float* __restrict__ coef, float* __restrict__ out_eta,
                              int k, int steps, int N) {
  if (threadIdx.x != 0 || blockIdx.x != 0) return;
  const float invN = 1.0f / (float)N;
  float A[6][7];
  for (int j = 0; j < 6; ++j) {
    for (int l = 0; l < 6; ++l) A[j][l] = (float)((j + 1) * (l + 1)) * xs[j + l] * invN;
    A[j][j] *= 1.001f;                       // REG = (1e-3, 0, 0)
    A[j][6]  = (float)(j + 1) * irs[j] * invN;
  }
  float eta[6];
  solve6(A, eta);
  for (int j = 0; j < 6; ++j) {
    coef[j] = eta[j];
    out_eta[k * 6 + j] = eta[j];
    if (k == steps - 1) out_eta[steps * 6 + j] = eta[j];  // duplicated last row
  }
}

// ---------------- corrector solve: theta = G(y)^-1 (bpe - moments(y)); dH; rows
__global__ void solve2_kernel(const float* __restrict__ ys, const float* __restrict__ irs,
                              const float* __restrict__ t, float* __restrict__ coef,
                              float* __restrict__ out_bpe, float* __restrict__ out_theta,
                              float* __restrict__ out_dH, int k, int steps, int N) {
  if (threadIdx.x != 0 || blockIdx.x != 0) return;
  const float invN = 1.0f / (float)N;
  const float h = t[k + 1] - t[k];
  float A[6][7], bpe[6];
  for (int j = 0; j < 6; ++j) {
    bpe[j] = irs[6 + j] * invN;
    for (int l = 0; l < 6; ++l) A[j][l] = (float)((j + 1) * (l + 1)) * ys[j + l] * invN;
    A[j][j] *= 1.001f;
    A[j][6]  = bpe[j] - ys[j + 1] * invN;
  }
  float th[6];
  solve6(A, th);
  const float inv_hs2 = 1.0f / (h * 0.25f);  // h * sigma^2, sigma = 0.5
  float dH = 0.0f;
  const bool last = (k == steps - 1);
  for (int j = 0; j < 6; ++j) {
    coef[j] = th[j];                          // raw theta drives corrector update
    const float to    = th[j] * inv_hs2;      // stored (scaled) theta
    const float dtphi = (float)(j + 1) * irs[12 + j] * invN;
    dH -= to * dtphi;
    out_theta[k * 6 + j] = to;
    out_bpe[(1 + k) * 6 + j] = bpe[j];
    if (last) {
      out_theta[steps * 6 + j] = to;
      out_bpe[(1 + steps) * 6 + j] = bpe[j];
    }
  }
  out_dH[k] = dH;
  if (last) out_dH[steps] = dH;
}

// ---------------- write moments row(s): S_{1..6}/N
__global__ void write_moments_kernel(const float* __restrict__ sums,
                                     float* __restrict__ dst1, float* __restrict__ dst2, int N) {
  const int j = threadIdx.x;
  if (j < 6) {
    const float m = sums[j + 1] / (float)N;
    dst1[j] = m;
    if (dst2) dst2[j] = m;
  }
}

// ---------------------------------------------------------------------------
extern "C" void kernel_launch(void* const* d_in, const int* in_sizes, int n_in,
                              void* d_out, int out_size, void* d_ws, size_t ws_size,
                              hipStream_t stream) {
  (void)n_in; (void)out_size; (void)ws_size;
  const float* x1 = (const float*)d_in[0];
  const float* x0 = (const float*)d_in[1];
  const float* xk = (const float*)d_in[2];
  const float* t  = (const float*)d_in[3];
  const int N = in_sizes[0];
  const int K = in_sizes[3];        // 33
  const int steps = K - 1;          // 32

  float* out = (float*)d_out;
  // return order: x_final(N), barphi_e(K+1,6), barphi_p(K+1,6),
  //               eta_all(K,6), theta_all(K,6), dH_all(K)
  const size_t O_be  = (size_t)N;
  const size_t O_bp  = O_be  + (size_t)(K + 1) * 6;
  const size_t O_eta = O_bp  + (size_t)(K + 1) * 6;
  const size_t O_th  = O_eta + (size_t)K * 6;
  const size_t O_dH  = O_th  + (size_t)K * 6;

  float* ws       = (float*)d_ws;
  float* ybuf     = ws;                        // N floats
  float* partials = ybuf + N;                  // NBLK*NACC floats
  float* xsums    = partials + (size_t)NBLK * NACC;  // 32
  float* ysums    = xsums + NACC;              // 32
  float* irsums   = ysums + NACC;              // 32
  float* coefA    = irsums + NACC;             // 8 (eta)
  float* coefB    = coefA + 8;                 // 8 (theta)
  float* zcoef    = coefB + 8;                 // 8 zeros
  float* xcur     = out;                       // working state == x_final slot

  zero8_kernel<<<1, 32, 0, stream>>>(zcoef);

  // barphi_p row 0 = moments(x_k); also copies x_k into xcur and seeds S(x)
  ur_kernel<<<NBLK, BLK, 0, stream>>>(xk, xcur, zcoef, t, 0, 0, N, partials);
  wmma_reduce_kernel<<<1, 32, 0, stream>>>(partials, xsums, NBLK);
  write_moments_kernel<<<1, 32, 0, stream>>>(xsums, out + O_bp, nullptr, N);

  // barphi_e row 0 = moments(x_0)  (copy lands in ybuf, overwritten later)
  ur_kernel<<<NBLK, BLK, 0, stream>>>(x0, ybuf, zcoef, t, 0, 0, N, partials);
  wmma_reduce_kernel<<<1, 32, 0, stream>>>(partials, ysums, NBLK);
  write_moments_kernel<<<1, 32, 0, stream>>>(ysums, out + O_be, nullptr, N);

  for (int k = 0; k < steps; ++k) {
    // interpolant sums (independent of x state)
    ir_kernel<<<NBLK, BLK, 0, stream>>>(x0, x1, t, k, N, partials);
    wmma_reduce_kernel<<<1, 32, 0, stream>>>(partials, irsums, NBLK);

    // eta from G(x) (S(x) was produced by previous corrector / init pass)
    solve1_kernel<<<1, 32, 0, stream>>>(xsums, irsums, coefA, out + O_eta, k, steps, N);

    // predictor: y = x + h*drift + noise ; S_m(y)
    ur_kernel<<<NBLK, BLK, 0, stream>>>(xcur, ybuf, coefA, t, k, 1, N, partials);
    wmma_reduce_kernel<<<1, 32, 0, stream>>>(partials, ysums, NBLK);

    // theta from G(y); writes theta/dH/bpe output rows
    solve2_kernel<<<1, 32, 0, stream>>>(ysums, irsums, t, coefB,
                                        out + O_be, out + O_th, out + O_dH, k, steps, N);

    // corrector: x' = y + gradphi(y)@theta ; S_m(x') feeds next step
    ur_kernel<<<NBLK, BLK, 0, stream>>>(ybuf, xcur, coefB, t, k, 0, N, partials);
    wmma_reduce_kernel<<<1, 32, 0, stream>>>(partials, xsums, NBLK);

    float* dst2 = (k == steps - 1) ? (out + O_bp + (size_t)(steps + 1) * 6) : nullptr;
    write_moments_kernel<<<1, 32, 0, stream>>>(xsums, out + O_bp + (size_t)(1 + k) * 6, dst2, N);
  }
}